// OHEMBCELoss_24103356465209
// MI455X (gfx1250) — compile-verified
//
#include <hip/hip_runtime.h>
#include <hip/hip_bf16.h>
#include <stdint.h>

#define BLK   256
#define TILE  4096          // u32 keys per TDM tile = 16 KB
#define BINS  256

typedef unsigned int       u32;
typedef unsigned long long u64;
typedef u32 v4u __attribute__((ext_vector_type(4)));
typedef int v8i __attribute__((ext_vector_type(8)));
typedef int v4i __attribute__((ext_vector_type(4)));

#if defined(__HIP_DEVICE_COMPILE__) && __has_builtin(__builtin_amdgcn_tensor_load_to_lds)
#define USE_TDM 1
#else
#define USE_TDM 0
#endif

#if defined(__HIP_DEVICE_COMPILE__) && __has_builtin(__builtin_amdgcn_s_wait_tensorcnt)
#define WAIT_TENSOR() __builtin_amdgcn_s_wait_tensorcnt(0)
#else
#define WAIT_TENSOR()
#endif

// ---------- helpers ----------
__device__ __forceinline__ float bce(float x, float t) {
  // max(x,0) - x*t + log1p(exp(-|x|)), numerically stable, always >= 0
  return fmaxf(x, 0.0f) - x * t + log1pf(__expf(-fabsf(x)));
}
// order-preserving float -> u32 (ascending)
__device__ __forceinline__ u32 key_of(float v) {
  u32 b = __float_as_uint(v);
  return (b & 0x80000000u) ? ~b : (b | 0x80000000u);
}
__device__ __forceinline__ float val_of(u32 k) {
  u32 b = (k & 0x80000000u) ? (k & 0x7fffffffu) : ~k;
  return __uint_as_float(b);
}

#if USE_TDM
// 1D TDM load of up to TILE u32 into LDS. nelem = valid elements (OOB reads -> 0).
// D# packing per CDNA5 ISA (group0: count/lds/global/type,
// group1: data_size=4B, tensor_dim0=nelem, tile_dim0=TILE, 1-D).
// amdgpu-toolchain (clang-23) 6-arg builtin: (v4u, v8i, v4i, v4i, v8i, i32 cpol).
__device__ __forceinline__ void tdm_load_u32(u32 lds_byte_off, const u32* gptr, u32 nelem) {
  unsigned long long ga = (unsigned long long)(uintptr_t)gptr;
  v4u g0;
  g0.x = 1u;                                              // count=1 valid descriptor
  g0.y = lds_byte_off;                                    // lds_addr
  g0.z = (u32)ga;                                         // global_addr[31:0]
  g0.w = ((u32)(ga >> 32) & 0x01FFFFFFu) | (2u << 30);    // global_addr[56:32] | type=2
  v8i g1;
  g1[0] = (int)(2u << 16);                                // data_size = 4 bytes
  g1[1] = (int)((nelem & 0xFFFFu) << 16);                 // tensor_dim0[15:0]
  g1[2] = (int)((nelem >> 16) & 0xFFFFu);                 // tensor_dim0[31:16], tensor_dim1=0
  g1[3] = (int)((u32)TILE << 16);                         // tile_dim0 = TILE
  g1[4] = 0;                                              // tile_dim1/2 unused
  g1[5] = 0; g1[6] = 0; g1[7] = 0;                        // strides
  v4i z4; z4[0] = z4[1] = z4[2] = z4[3] = 0;              // group2/3 (dim2/3: unused)
  v8i z8; z8[0] = z8[1] = z8[2] = z8[3] = z8[4] = z8[5] = z8[6] = z8[7] = 0;
  __builtin_amdgcn_tensor_load_to_lds(g0, g1, z4, z4, z8, 0);
}
#endif

// ---------- workspace layout (in d_ws) ----------
// [0 .. 1023]   : u32 hist[256]
// [1024 ..]     : state: st[0]=prefix/T, st[1]=krem, st[2..3]=u64 fixed-point sum,
//                 st[4]=count(key>T)
// [4096 ..]     : u32 keys[n]   (only if ws_size large enough)

__global__ void init_ws(u32* __restrict__ ghist, u32* __restrict__ st) {
  int t = threadIdx.x;
  ghist[t] = 0;
  if (t < 8) st[t] = 0;
}

// ---------- pass 1: compute loss keys, store, histogram top 8 bits ----------
__global__ void pass1_keys_hist(const float4* __restrict__ lg, const float4* __restrict__ tg,
                                uint4* __restrict__ keys, u32* __restrict__ ghist, long long n4) {
  __shared__ u32 lhist[BINS];
  for (int i = threadIdx.x; i < BINS; i += BLK) lhist[i] = 0;
  __syncthreads();
  long long stride = (long long)gridDim.x * BLK;
  for (long long i = (long long)blockIdx.x * BLK + threadIdx.x; i < n4; i += stride) {
    if (i + stride < n4) {
      __builtin_prefetch(&lg[i + stride], 0, 0);   // global_prefetch_b8
      __builtin_prefetch(&tg[i + stride], 0, 0);
    }
    float4 x = lg[i], t = tg[i];
    u32 k0 = key_of(bce(x.x, t.x));
    u32 k1 = key_of(bce(x.y, t.y));
    u32 k2 = key_of(bce(x.z, t.z));
    u32 k3 = key_of(bce(x.w, t.w));
    keys[i] = make_uint4(k0, k1, k2, k3);
    atomicAdd(&lhist[k0 >> 24], 1u);
    atomicAdd(&lhist[k1 >> 24], 1u);
    atomicAdd(&lhist[k2 >> 24], 1u);
    atomicAdd(&lhist[k3 >> 24], 1u);
  }
  __syncthreads();
  for (int i = threadIdx.x; i < BINS; i += BLK) atomicAdd(&ghist[i], lhist[i]);
}

// ---------- refinement histogram over keys (TDM double-buffered through LDS) ----------
__global__ void hist_keys_kernel(const u32* __restrict__ keys, u32* __restrict__ ghist,
                                 const u32* __restrict__ st, int shift, u32 mask, long long n) {
  __shared__ u32 lhist[BINS];
  for (int i = threadIdx.x; i < BINS; i += BLK) lhist[i] = 0;
  u32 prefix = st[0];
  __syncthreads();
#if USE_TDM
  __shared__ __align__(16) u32 tile[2][TILE];
  long long ntiles = (n + TILE - 1) / TILE;
  long long t0 = blockIdx.x;
  if (t0 < ntiles && threadIdx.x < 32) {
    long long rem = n - t0 * TILE;
    tdm_load_u32((u32)(uintptr_t)&tile[0][0], keys + t0 * TILE,
                 (u32)(rem < TILE ? rem : TILE));
  }
  int cur = 0;
  for (long long t = t0; t < ntiles; t += gridDim.x) {
    if (threadIdx.x < 32) WAIT_TENSOR();
    __syncthreads();                       // tile[cur] ready; prev buffer fully consumed
    long long tn = t + gridDim.x;
    if (tn < ntiles && threadIdx.x < 32) {
      long long rem = n - tn * TILE;
      tdm_load_u32((u32)(uintptr_t)&tile[cur ^ 1][0], keys + tn * TILE,
                   (u32)(rem < TILE ? rem : TILE));
    }
    long long remc = n - t * TILE;
    u32 cnt = (u32)(remc < TILE ? remc : TILE);
    u32 cnt4 = cnt >> 2;
    const uint4* t4 = (const uint4*)&tile[cur][0];
    for (u32 i = threadIdx.x; i < cnt4; i += BLK) {        // ds_load_b128
      uint4 kv = t4[i];
      if ((kv.x & mask) == prefix) atomicAdd(&lhist[(kv.x >> shift) & 0xFF], 1u);
      if ((kv.y & mask) == prefix) atomicAdd(&lhist[(kv.y >> shift) & 0xFF], 1u);
      if ((kv.z & mask) == prefix) atomicAdd(&lhist[(kv.z >> shift) & 0xFF], 1u);
      if ((kv.w & mask) == prefix) atomicAdd(&lhist[(kv.w >> shift) & 0xFF], 1u);
    }
    for (u32 i = (cnt4 << 2) + threadIdx.x; i < cnt; i += BLK) {   // tail (unused here)
      u32 kk = tile[cur][i];
      if ((kk & mask) == prefix) atomicAdd(&lhist[(kk >> shift) & 0xFF], 1u);
    }
    cur ^= 1;
  }
#else
  long long n4 = n / 4;
  long long stride = (long long)gridDim.x * BLK;
  for (long long i = (long long)blockIdx.x * BLK + threadIdx.x; i < n4; i += stride) {
    uint4 kv = ((const uint4*)keys)[i];
    if ((kv.x & mask) == prefix) atomicAdd(&lhist[(kv.x >> shift) & 0xFF], 1u);
    if ((kv.y & mask) == prefix) atomicAdd(&lhist[(kv.y >> shift) & 0xFF], 1u);
    if ((kv.z & mask) == prefix) atomicAdd(&lhist[(kv.z >> shift) & 0xFF], 1u);
    if ((kv.w & mask) == prefix) atomicAdd(&lhist[(kv.w >> shift) & 0xFF], 1u);
  }
#endif
  __syncthreads();
  for (int i = threadIdx.x; i < BINS; i += BLK) atomicAdd(&ghist[i], lhist[i]);
}

// ---------- scan one radix level: pick digit, update prefix/krem, zero hist ----------
__global__ void scan_level(u32* __restrict__ ghist, u32* __restrict__ st, int shift, u32 kinit) {
  __shared__ u32 h[BINS];
  int t = threadIdx.x;
  h[t] = ghist[t];
  ghist[t] = 0;                       // ready for next level
  __syncthreads();
  if (t == 0) {
    u32 krem   = (shift == 24) ? kinit : st[1];
    u32 prefix = (shift == 24) ? 0u    : st[0];
    u32 cum = 0; int d = 0;
    for (int i = 255; i >= 0; --i) {  // descending (top-k = largest)
      u32 c = h[i];
      if (cum + c >= krem) { d = i; break; }
      cum += c;
    }
    st[0] = prefix | ((u32)d << (u32)shift);
    st[1] = krem - cum;
  }
}

// ---------- final sum of elements strictly above threshold (deterministic fixed point) ----------
__device__ __forceinline__ void acc_key(u32 kk, u32 T, u64& lsum, u32& lcnt) {
  if (kk > T) { lsum += (u64)((double)val_of(kk) * 4294967296.0); lcnt++; }
}

__global__ void sum_keys_kernel(const u32* __restrict__ keys, u32* __restrict__ st, long long n) {
  __shared__ u64 red[BLK];
  __shared__ u32 redc[BLK];
  u32 T = st[0];
  u64 lsum = 0; u32 lcnt = 0;
#if USE_TDM
  __shared__ __align__(16) u32 tile[2][TILE];
  long long ntiles = (n + TILE - 1) / TILE;
  long long t0 = blockIdx.x;
  if (t0 < ntiles && threadIdx.x < 32) {
    long long rem = n - t0 * TILE;
    tdm_load_u32((u32)(uintptr_t)&tile[0][0], keys + t0 * TILE,
                 (u32)(rem < TILE ? rem : TILE));
  }
  int cur = 0;
  for (long long t = t0; t < ntiles; t += gridDim.x) {
    if (threadIdx.x < 32) WAIT_TENSOR();
    __syncthreads();
    long long tn = t + gridDim.x;
    if (tn < ntiles && threadIdx.x < 32) {
      long long rem = n - tn * TILE;
      tdm_load_u32((u32)(uintptr_t)&tile[cur ^ 1][0], keys + tn * TILE,
                   (u32)(rem < TILE ? rem : TILE));
    }
    long long remc = n - t * TILE;
    u32 cnt = (u32)(remc < TILE ? remc : TILE);
    u32 cnt4 = cnt >> 2;
    const uint4* t4 = (const uint4*)&tile[cur][0];
    for (u32 i = threadIdx.x; i < cnt4; i += BLK) {        // ds_load_b128
      uint4 kv = t4[i];
      acc_key(kv.x, T, lsum, lcnt);
      acc_key(kv.y, T, lsum, lcnt);
      acc_key(kv.z, T, lsum, lcnt);
      acc_key(kv.w, T, lsum, lcnt);
    }
    for (u32 i = (cnt4 << 2) + threadIdx.x; i < cnt; i += BLK)     // tail (unused here)
      acc_key(tile[cur][i], T, lsum, lcnt);
    cur ^= 1;
  }
#else
  long long n4 = n / 4;
  long long stride = (long long)gridDim.x * BLK;
  for (long long i = (long long)blockIdx.x * BLK + threadIdx.x; i < n4; i += stride) {
    uint4 kv = ((const uint4*)keys)[i];
    acc_key(kv.x, T, lsum, lcnt);
    acc_key(kv.y, T, lsum, lcnt);
    acc_key(kv.z, T, lsum, lcnt);
    acc_key(kv.w, T, lsum, lcnt);
  }
#endif
  red[threadIdx.x] = lsum; redc[threadIdx.x] = lcnt;
  __syncthreads();
  for (int off = BLK / 2; off > 0; off >>= 1) {
    if (threadIdx.x < off) { red[threadIdx.x] += red[threadIdx.x + off];
                             redc[threadIdx.x] += redc[threadIdx.x + off]; }
    __syncthreads();
  }
  if (threadIdx.x == 0) {
    atomicAdd((unsigned long long*)(st + 2), red[0]);   // integer: order-independent exact
    atomicAdd(&st[4], redc[0]);
  }
}

// ---------- fused fallbacks (used when workspace can't hold the key array) ----------
__global__ void hist_fused(const float4* __restrict__ lg, const float4* __restrict__ tg,
                           u32* __restrict__ ghist, const u32* __restrict__ st,
                           int shift, u32 mask, long long n4) {
  __shared__ u32 lhist[BINS];
  for (int i = threadIdx.x; i < BINS; i += BLK) lhist[i] = 0;
  u32 prefix = st[0];
  __syncthreads();
  long long stride = (long long)gridDim.x * BLK;
  for (long long i = (long long)blockIdx.x * BLK + threadIdx.x; i < n4; i += stride) {
    if (i + stride < n4) { __builtin_prefetch(&lg[i + stride], 0, 0);
                           __builtin_prefetch(&tg[i + stride], 0, 0); }
    float4 x = lg[i], t = tg[i];
    u32 kk;
    kk = key_of(bce(x.x, t.x)); if ((kk & mask) == prefix) atomicAdd(&lhist[(kk >> shift) & 0xFF], 1u);
    kk = key_of(bce(x.y, t.y)); if ((kk & mask) == prefix) atomicAdd(&lhist[(kk >> shift) & 0xFF], 1u);
    kk = key_of(bce(x.z, t.z)); if ((kk & mask) == prefix) atomicAdd(&lhist[(kk >> shift) & 0xFF], 1u);
    kk = key_of(bce(x.w, t.w)); if ((kk & mask) == prefix) atomicAdd(&lhist[(kk >> shift) & 0xFF], 1u);
  }
  __syncthreads();
  for (int i = threadIdx.x; i < BINS; i += BLK) atomicAdd(&ghist[i], lhist[i]);
}

__global__ void sum_fused(const float4* __restrict__ lg, const float4* __restrict__ tg,
                          u32* __restrict__ st, long long n4) {
  __shared__ u64 red[BLK];
  __shared__ u32 redc[BLK];
  u32 T = st[0];
  u64 lsum = 0; u32 lcnt = 0;
  long long stride = (long long)gridDim.x * BLK;
  for (long long i = (long long)blockIdx.x * BLK + threadIdx.x; i < n4; i += stride) {
    float4 x = lg[i], t = tg[i];
    acc_key(key_of(bce(x.x, t.x)), T, lsum, lcnt);
    acc_key(key_of(bce(x.y, t.y)), T, lsum, lcnt);
    acc_key(key_of(bce(x.z, t.z)), T, lsum, lcnt);
    acc_key(key_of(bce(x.w, t.w)), T, lsum, lcnt);
  }
  red[threadIdx.x] = lsum; redc[threadIdx.x] = lcnt;
  __syncthreads();
  for (int off = BLK / 2; off > 0; off >>= 1) {
    if (threadIdx.x < off) { red[threadIdx.x] += red[threadIdx.x + off];
                             redc[threadIdx.x] += redc[threadIdx.x + off]; }
    __syncthreads();
  }
  if (threadIdx.x == 0) {
    atomicAdd((unsigned long long*)(st + 2), red[0]);
    atomicAdd(&st[4], redc[0]);
  }
}

__global__ void finalize_kernel(const u32* __restrict__ st, float* __restrict__ out, long long k) {
  u32 T = st[0];
  u64 sum = *(const u64*)(st + 2);
  u32 cgt = st[4];
  double s = (double)sum * (1.0 / 4294967296.0);
  double mean = (s + (double)(k - (long long)cgt) * (double)val_of(T)) / (double)k;
  out[0] = (float)mean;
}

extern "C" void kernel_launch(void* const* d_in, const int* in_sizes, int n_in,
                              void* d_out, int out_size, void* d_ws, size_t ws_size,
                              hipStream_t stream) {
  const float* logits  = (const float*)d_in[0];
  const float* targets = (const float*)d_in[1];
  long long n  = (long long)in_sizes[0];          // 8192*4096 = 33,554,432
  long long n4 = n / 4;                           // divisible here
  long long k  = (n + 1) / 2;                     // int(n*0.5 + 0.5)

  u32* ghist = (u32*)d_ws;
  u32* st    = (u32*)((char*)d_ws + 1024);
  u32* keys  = (u32*)((char*)d_ws + 4096);
  bool keys_mode = ws_size >= (size_t)n * 4 + 4096;

  const int G1 = 2048;   // streaming compute pass
  const int G2 = 1024;   // L2-resident key passes (8192 tiles / 1024 blocks = 8 tiles each)

  init_ws<<<1, 256, 0, stream>>>(ghist, st);

  if (keys_mode) {
    pass1_keys_hist<<<G1, BLK, 0, stream>>>((const float4*)logits, (const float4*)targets,
                                            (uint4*)keys, ghist, n4);
    scan_level<<<1, 256, 0, stream>>>(ghist, st, 24, (u32)k);
    const int shifts[3] = {16, 8, 0};
    for (int i = 0; i < 3; ++i) {
      int s = shifts[i];
      u32 mask = 0xFFFFFFFFu << (s + 8);
      hist_keys_kernel<<<G2, BLK, 0, stream>>>(keys, ghist, st, s, mask, n);
      scan_level<<<1, 256, 0, stream>>>(ghist, st, s, (u32)k);
    }
    sum_keys_kernel<<<G2, BLK, 0, stream>>>(keys, st, n);
  } else {
    const int shifts[4] = {24, 16, 8, 0};
    for (int i = 0; i < 4; ++i) {
      int s = shifts[i];
      u32 mask = (s == 24) ? 0u : (0xFFFFFFFFu << (s + 8));
      hist_fused<<<G1, BLK, 0, stream>>>((const float4*)logits, (const float4*)targets,
                                         ghist, st, s, mask, n4);
      scan_level<<<1, 256, 0, stream>>>(ghist, st, s, (u32)k);
    }
    sum_fused<<<G1, BLK, 0, stream>>>((const float4*)logits, (const float4*)targets, st, n4);
  }
  finalize_kernel<<<1, 1, 0, stream>>>(st, (float*)d_out, k);
}